// ChebConvGAD_36043365548319
// MI455X (gfx1250) — compile-verified
//
#include <hip/hip_runtime.h>
#include <hip/hip_bf16.h>

typedef __attribute__((ext_vector_type(2))) float v2f;
typedef __attribute__((ext_vector_type(8))) float v8f;

#define N_NODES 50000
#define N_EDGES 800000
#define KPAD    196          // max K (192) + pad to dodge LDS bank conflicts

// -------------------------------------------------------------------------
// small utility kernels
// -------------------------------------------------------------------------
__global__ void zero_f32(float* __restrict__ p, int n) {
    int i = blockIdx.x * blockDim.x + threadIdx.x;
    if (i < n) p[i] = 0.0f;
}

__global__ void deg_accum(const int* __restrict__ dst, float* __restrict__ deg, int e) {
    int i = blockIdx.x * blockDim.x + threadIdx.x;
    if (i < e) atomicAdd(&deg[dst[i]], 1.0f);
}

__global__ void deg_to_dinv(float* __restrict__ deg, int n) {
    int i = blockIdx.x * blockDim.x + threadIdx.x;
    if (i < n) deg[i] = rsqrtf(fmaxf(deg[i], 1.0f));
}

// copy h [N,64] into xcat column block 0 (row stride 192)
__global__ void copy_to_xcat(const float* __restrict__ h, float* __restrict__ xcat, int n64) {
    int i = blockIdx.x * blockDim.x + threadIdx.x;
    if (i >= n64) return;
    int node = i >> 6, f = i & 63;
    xcat[node * 192 + f] = h[i];
}

// -------------------------------------------------------------------------
// SpMM: agg[dst] += x[src] * dinv[src]  (thread = (edge, feature))
// -------------------------------------------------------------------------
__global__ void spmm_edges(const int* __restrict__ src, const int* __restrict__ dst,
                           const float* __restrict__ x, int xStride,
                           const float* __restrict__ dinv,
                           float* __restrict__ agg, int e) {
    long long i = (long long)blockIdx.x * blockDim.x + threadIdx.x;
    if (i >= (long long)e * 64) return;
    int eidx = (int)(i >> 6);
    int f    = (int)(i & 63);
    int s = src[eidx], d = dst[eidx];
    float v = x[s * xStride + f] * dinv[s];
    atomicAdd(&agg[d * 64 + f], v);
}

// X1 = cAgg * (dinv ⊙ agg) + cX0 * X0      (X0 at xcat col 0, write col 64)
__global__ void cheb_x1(const float* __restrict__ agg, float* __restrict__ xcat,
                        const float* __restrict__ dinv, float cAgg, float cX0, int n64) {
    int i = blockIdx.x * blockDim.x + threadIdx.x;
    if (i >= n64) return;
    int node = i >> 6, f = i & 63;
    float a = agg[i] * dinv[node];
    xcat[node * 192 + 64 + f] = cAgg * a + cX0 * xcat[node * 192 + f];
}

// X2 = cAgg * (dinv ⊙ agg) + cX1 * X1 + cX0 * X0   (write col 128)
__global__ void cheb_x2(const float* __restrict__ agg, float* __restrict__ xcat,
                        const float* __restrict__ dinv,
                        float cAgg, float cX1, float cX0, int n64) {
    int i = blockIdx.x * blockDim.x + threadIdx.x;
    if (i >= n64) return;
    int node = i >> 6, f = i & 63;
    float a  = agg[i] * dinv[node];
    float x0 = xcat[node * 192 + f];
    float x1 = xcat[node * 192 + 64 + f];
    xcat[node * 192 + 128 + f] = cAgg * a + cX1 * x1 + cX0 * x0;
}

// -------------------------------------------------------------------------
// WMMA fp32 GEMM: Out[N_NODES, Ncols] = act(X[N_NODES, Kdim] * W^T + bias)
//   W is [Ncols, Kdim] row-major (PyTorch Linear layout).
//   128 threads = 4 wave32; block tile = 64 rows x 16 cols (blockIdx.y -> col tile)
//   Each wave: one 16x16 C tile, K-loop of v_wmma_f32_16x16x4_f32.
// -------------------------------------------------------------------------
__global__ void __launch_bounds__(128)
gemm_wmma(const float* __restrict__ X, int xStride,
          const float* __restrict__ W, const float* __restrict__ bias,
          float* __restrict__ Out, int outStride,
          int Kdim, int Ncols, int doRelu) {
    __shared__ float sW[16 * KPAD];        // W tile: 16 output cols x Kdim
    __shared__ float sA[4][16 * KPAD];     // per-wave A tiles: 16 rows x Kdim

    const int tid  = threadIdx.x;
    const int wave = tid >> 5;
    const int lane = tid & 31;
    const int rowBase = blockIdx.x * 64;
    const int waveRow = rowBase + wave * 16;
    const int n0 = blockIdx.y * 16;

    // stage W tile (shared by all 4 waves), coalesced on k
    for (int idx = tid; idx < 16 * Kdim; idx += 128) {
        int n = idx / Kdim, k = idx - n * Kdim;
        int gn = n0 + n;
        sW[n * KPAD + k] = (gn < Ncols) ? W[gn * Kdim + k] : 0.0f;
    }
    // stage this wave's 16-row A tile, coalesced on k
    for (int idx = lane; idx < 16 * Kdim; idx += 32) {
        int r = idx / Kdim, k = idx - r * Kdim;
        int row = waveRow + r;
        sA[wave][r * KPAD + k] = (row < N_NODES) ? X[row * xStride + k] : 0.0f;
    }
    __syncthreads();

    const int half = lane >> 4;   // 0: lanes 0-15, 1: lanes 16-31
    const int l16  = lane & 15;

    v8f c = {};
    const float* aRow = &sA[wave][0];
    for (int k0 = 0; k0 < Kdim; k0 += 4) {
        // ISA 7.12.2: 32-bit A 16x4 -> vgpr j, lane-group g holds K = k0 + 2g + j
        int ka = k0 + 2 * half;
        v2f a, b;
        a.x = aRow[l16 * KPAD + ka];
        a.y = aRow[l16 * KPAD + ka + 1];
        // B[k][n] = W[n0+n][k], n = lane%16, same K striping as A
        b.x = sW[l16 * KPAD + ka];
        b.y = sW[l16 * KPAD + ka + 1];
        c = __builtin_amdgcn_wmma_f32_16x16x4_f32(
                /*neg_a=*/false, a, /*neg_b=*/false, b,
                /*c_mod=*/(short)0, c, /*reuse_a=*/false, /*reuse_b=*/false);
    }

    // C/D layout: vgpr r, lane l -> M = r + 8*(l/16), N = l%16
    int col = n0 + l16;
    if (col < Ncols) {
        float bv = bias[col];
        #pragma unroll
        for (int r = 0; r < 8; ++r) {
            int row = waveRow + r + 8 * half;
            if (row < N_NODES) {
                float v = c[r] + bv;
                if (doRelu) v = fmaxf(v, 0.0f);
                Out[row * outStride + col] = v;
            }
        }
    }
}

// -------------------------------------------------------------------------
// host-side launch
// -------------------------------------------------------------------------
extern "C" void kernel_launch(void* const* d_in, const int* in_sizes, int n_in,
                              void* d_out, int out_size, void* d_ws, size_t ws_size,
                              hipStream_t stream) {
    const float* in_feat = (const float*)d_in[0];
    const int*   src     = (const int*)d_in[1];
    const int*   dst     = (const int*)d_in[2];
    const float* W1 = (const float*)d_in[3];   const float* b1 = (const float*)d_in[4];
    const float* W2 = (const float*)d_in[5];   const float* b2 = (const float*)d_in[6];
    const float* Wc1 = (const float*)d_in[7];  const float* bc1 = (const float*)d_in[8];
    const float* Wc2 = (const float*)d_in[9];  const float* bc2 = (const float*)d_in[10];
    const float* W3 = (const float*)d_in[11];  const float* b3 = (const float*)d_in[12];
    const float* W4 = (const float*)d_in[13];  const float* b4 = (const float*)d_in[14];

    // workspace layout (bytes): dinv | bufA | bufB | xcat | agg
    char* ws = (char*)d_ws;
    const size_t SZ_H   = (size_t)N_NODES * 64 * 4;   // 12.8 MB
    const size_t SZ_CAT = (size_t)N_NODES * 192 * 4;  // 38.4 MB
    float* dinv = (float*)(ws);
    float* bufA = (float*)(ws + 200192);
    float* bufB = (float*)(ws + 200192 + SZ_H);
    float* xcat = (float*)(ws + 200192 + 2 * SZ_H);
    float* agg  = (float*)(ws + 200192 + 2 * SZ_H + SZ_CAT);

    const int n64 = N_NODES * 64;
    dim3 blkS(256);
    dim3 gN((N_NODES + 255) / 256), gE((N_EDGES + 255) / 256), gNF((n64 + 255) / 256);
    long long espmm = (long long)N_EDGES * 64;
    dim3 gSP((unsigned)((espmm + 255) / 256));

    // degrees -> d_invsqrt (in place)
    zero_f32<<<gN, blkS, 0, stream>>>(dinv, N_NODES);
    deg_accum<<<gE, blkS, 0, stream>>>(dst, dinv, N_EDGES);
    deg_to_dinv<<<gN, blkS, 0, stream>>>(dinv, N_NODES);

    dim3 gemmBlk(128);
    dim3 gG64((N_NODES + 63) / 64, 4);   // 64 output cols
    dim3 gG2((N_NODES + 63) / 64, 1);    // 2 output cols

    // h = relu(in_feat @ W1^T + b1); h = relu(h @ W2^T + b2)
    gemm_wmma<<<gG64, gemmBlk, 0, stream>>>(in_feat, 128, W1, b1, bufA, 64, 128, 64, 1);
    gemm_wmma<<<gG64, gemmBlk, 0, stream>>>(bufA,     64, W2, b2, bufB, 64,  64, 64, 1);

    const float rn = 2.0f / 2.0f;  // 2 / LAMBDA_MAX
    const float* h_in = bufB;
    float*       h_out = bufA;
    const float* Wc[2] = { Wc1, Wc2 };
    const float* bc[2] = { bc1, bc2 };
    for (int layer = 0; layer < 2; ++layer) {
        // X0 -> xcat[:,0:64]
        copy_to_xcat<<<gNF, blkS, 0, stream>>>(h_in, xcat, n64);
        // X1 = -rn * a_norm(X0) + (rn-1) * X0
        zero_f32<<<gNF, blkS, 0, stream>>>(agg, n64);
        spmm_edges<<<gSP, blkS, 0, stream>>>(src, dst, xcat, 192, dinv, agg, N_EDGES);
        cheb_x1<<<gNF, blkS, 0, stream>>>(agg, xcat, dinv, -rn, rn - 1.0f, n64);
        // X2 = -2rn * a_norm(X1) + 2(rn-1) * X1 - X0
        zero_f32<<<gNF, blkS, 0, stream>>>(agg, n64);
        spmm_edges<<<gSP, blkS, 0, stream>>>(src, dst, xcat + 64, 192, dinv, agg, N_EDGES);
        cheb_x2<<<gNF, blkS, 0, stream>>>(agg, xcat, dinv, -2.0f * rn, 2.0f * (rn - 1.0f), -1.0f, n64);
        // relu(concat(X0,X1,X2) @ Wc^T + bc)
        gemm_wmma<<<gG64, gemmBlk, 0, stream>>>(xcat, 192, Wc[layer], bc[layer], h_out, 64, 192, 64, 1);
        const float* t = h_in; h_in = h_out; h_out = (float*)t;  // swap bufA/bufB
    }
    // h_in now holds cheb2 output (bufB after two swaps)
    gemm_wmma<<<gG64, gemmBlk, 0, stream>>>(h_in, 64, W3, b3, (float*)h_out, 64, 64, 64, 1);
    gemm_wmma<<<gG2,  gemmBlk, 0, stream>>>((const float*)h_out, 64, W4, b4, (float*)d_out, 2, 64, 2, 0);
}